// LightGCN_38457137168961
// MI455X (gfx1250) — compile-verified
//
#include <hip/hip_runtime.h>

#define EMB 128

typedef __attribute__((ext_vector_type(2))) float v2f;
typedef __attribute__((ext_vector_type(8))) float v8f;

// ---------------------------------------------------------------------------
// Zero the h1/h2/h3 scratch buffers (float4-vectorized).
// ---------------------------------------------------------------------------
__global__ void lgcn_zero_ws(float4* __restrict__ p, unsigned long long n4) {
    unsigned long long i = (unsigned long long)blockIdx.x * blockDim.x + threadIdx.x;
    if (i < n4) p[i] = make_float4(0.f, 0.f, 0.f, 0.f);
}

// ---------------------------------------------------------------------------
// SpMM scatter: one wave (32 lanes) per nonzero edge. Each lane handles a
// float4 chunk of the 128-dim row: coalesced 512B gather of src[col], then
// 4 returnless global_atomic_add_f32 into hout[row]. Both src and dst live
// in the 192MB L2 (feature matrix is 61.4MB), so this is L2-throughput bound.
// srcA/srcB + split implements the layer-0 "ego" concat without a copy:
//   node c < split -> srcA[c], else srcB[c - split].
// ---------------------------------------------------------------------------
__global__ void lgcn_spmm_scatter(const int* __restrict__ rowv,
                                  const int* __restrict__ colv,
                                  const float* __restrict__ valv,
                                  const float* __restrict__ srcA,
                                  const float* __restrict__ srcB,
                                  int split,
                                  float* __restrict__ hout,
                                  int nnz) {
    unsigned tid  = blockIdx.x * blockDim.x + threadIdx.x;
    unsigned e    = tid >> 5;
    unsigned lane = tid & 31;
    if (e >= (unsigned)nnz) return;

    // Stream-prefetch upcoming edge metadata (global_prefetch_b8).
    if (lane == 0 && (e + 4096) < (unsigned)nnz) {
        __builtin_prefetch(colv + e + 4096, 0, 1);
        __builtin_prefetch(rowv + e + 4096, 0, 1);
        __builtin_prefetch(valv + e + 4096, 0, 1);
    }

    int   r = rowv[e];
    int   c = colv[e];
    float v = valv[e];

    const float* src = (c < split) ? (srcA + (size_t)c * EMB)
                                   : (srcB + (size_t)(c - split) * EMB);
    float4 g = *reinterpret_cast<const float4*>(src + (lane << 2));

    float* dst = hout + (size_t)r * EMB + (lane << 2);
    (void)__hip_atomic_fetch_add(dst + 0, v * g.x, __ATOMIC_RELAXED, __HIP_MEMORY_SCOPE_AGENT);
    (void)__hip_atomic_fetch_add(dst + 1, v * g.y, __ATOMIC_RELAXED, __HIP_MEMORY_SCOPE_AGENT);
    (void)__hip_atomic_fetch_add(dst + 2, v * g.z, __ATOMIC_RELAXED, __HIP_MEMORY_SCOPE_AGENT);
    (void)__hip_atomic_fetch_add(dst + 3, v * g.w, __ATOMIC_RELAXED, __HIP_MEMORY_SCOPE_AGENT);
}

// ---------------------------------------------------------------------------
// Finalize item embeddings with V_WMMA_F32_16X16X4_F32:
//   final[f] = 0.25 * (item_emb[f] + h1[f] + h2[f] + h3[f])
// One wave computes 16 consecutive elements: the four layer terms are loaded
// into the B matrix (4x16 f32, 2 VGPRs: lanes 0-15 carry two K-rows, lanes
// 16-31 the other two), A is the constant 0.25 matrix. Every column of
// D = A*B equals 0.25 * sum_k B[k,n]; since all K rows have equal weight the
// result is invariant to the exact K-row-to-slot permutation of the B layout.
// Lanes 0-15 store column n = lane (coalesced 64B store).
// Guard is wave-uniform so EXEC is all-ones at the WMMA (ISA requirement).
// ---------------------------------------------------------------------------
__global__ void lgcn_finalize_items_wmma(const float* __restrict__ item_emb,
                                         const float* __restrict__ h1,
                                         const float* __restrict__ h2,
                                         const float* __restrict__ h3,
                                         unsigned long long itemsOff,
                                         unsigned long long itemElems,
                                         float* __restrict__ out) {
    unsigned tid  = blockIdx.x * blockDim.x + threadIdx.x;
    unsigned wave = tid >> 5;
    unsigned lane = tid & 31;
    if ((unsigned long long)wave * 16ull >= itemElems) return;  // wave-uniform

    unsigned long long f = (unsigned long long)wave * 16ull + (lane & 15);

    float s0, s1;
    if (lane < 16) {
        s0 = item_emb[f];
        s1 = h1[itemsOff + f];
    } else {
        s0 = h2[itemsOff + f];
        s1 = h3[itemsOff + f];
    }

    v2f a; a[0] = 0.25f; a[1] = 0.25f;   // A = all 0.25 (layout-independent)
    v2f b; b[0] = s0;    b[1] = s1;      // B = the four layer terms
    v8f c = {};
    c = __builtin_amdgcn_wmma_f32_16x16x4_f32(
            /*neg_a=*/false, a, /*neg_b=*/false, b,
            /*c_mod=*/(short)0, c, /*reuse_a=*/false, /*reuse_b=*/false);

    if (lane < 16) out[f] = c[0];
}

// ---------------------------------------------------------------------------
// Batched gather: out[b,:] = 0.25*(base_emb[idx[b],:] + h1/h2/h3[node,:])
// where node = idx[b] + nodeOff. One 128-thread block per batch element.
// ---------------------------------------------------------------------------
__global__ void lgcn_gather_final(const int* __restrict__ idx,
                                  const float* __restrict__ base_emb,
                                  const float* __restrict__ h1,
                                  const float* __restrict__ h2,
                                  const float* __restrict__ h3,
                                  int nodeOff,
                                  float* __restrict__ out) {
    int b = blockIdx.x;
    int d = threadIdx.x;
    int i = idx[b];
    size_t eoff = (size_t)i * EMB + d;
    size_t noff = (size_t)(i + nodeOff) * EMB + d;
    out[(size_t)b * EMB + d] =
        0.25f * (base_emb[eoff] + h1[noff] + h2[noff] + h3[noff]);
}

extern "C" void kernel_launch(void* const* d_in, const int* in_sizes, int n_in,
                              void* d_out, int out_size, void* d_ws, size_t ws_size,
                              hipStream_t stream) {
    const int*   users = (const int*)d_in[0];
    const int*   pos   = (const int*)d_in[1];
    const int*   neg   = (const int*)d_in[2];
    const float* uemb  = (const float*)d_in[3];
    const float* iemb  = (const float*)d_in[4];
    const int*   arow  = (const int*)d_in[5];
    const int*   acol  = (const int*)d_in[6];
    const float* aval  = (const float*)d_in[7];
    // n_layers (d_in[8]) is the compile-time constant 3 in the reference.

    const int batch     = in_sizes[0];
    const int num_users = in_sizes[3] / EMB;
    const int num_items = in_sizes[4] / EMB;
    const int n_nodes   = num_users + num_items;
    const int nnz       = in_sizes[5];

    // Workspace layout: h1 | h2 | h3, each n_nodes*EMB floats (61.44 MB).
    const size_t bufElems = (size_t)n_nodes * EMB;
    float* h1 = (float*)d_ws;
    float* h2 = h1 + bufElems;
    float* h3 = h2 + bufElems;

    // 1) Zero the three propagation buffers (atomic scatter targets).
    {
        unsigned long long n4 = (3ull * bufElems) / 4ull;
        unsigned blocks = (unsigned)((n4 + 255ull) / 256ull);
        lgcn_zero_ws<<<blocks, 256, 0, stream>>>((float4*)d_ws, n4);
    }

    // 2) Three SpMM layers: h1 = A*ego, h2 = A*h1, h3 = A*h2.
    {
        unsigned blocks = (unsigned)(((size_t)nnz * 32 + 255) / 256);
        lgcn_spmm_scatter<<<blocks, 256, 0, stream>>>(
            arow, acol, aval, uemb, iemb, num_users, h1, nnz);  // ego split-gather
        lgcn_spmm_scatter<<<blocks, 256, 0, stream>>>(
            arow, acol, aval, h1, h1, n_nodes, h2, nnz);
        lgcn_spmm_scatter<<<blocks, 256, 0, stream>>>(
            arow, acol, aval, h2, h2, n_nodes, h3, nnz);
    }

    float* out   = (float*)d_out;
    float* u_out = out;
    float* p_out = out + (size_t)batch * EMB;
    float* n_out = out + 2ull * (size_t)batch * EMB;
    float* i_out = out + 3ull * (size_t)batch * EMB;  // full i_g (40000 x 128)

    // 3) WMMA finalize of the full item block -> d_out tail region.
    {
        unsigned long long itemsOff  = (unsigned long long)num_users * EMB;
        unsigned long long itemElems = (unsigned long long)num_items * EMB;
        unsigned long long threads   = itemElems * 2ull;  // 32 threads per 16 elems
        unsigned blocks = (unsigned)((threads + 255ull) / 256ull);
        lgcn_finalize_items_wmma<<<blocks, 256, 0, stream>>>(
            iemb, h1, h2, h3, itemsOff, itemElems, i_out);
    }

    // 4) Batched gathers: u_g[users], i_g[pos_items], i_g[neg_items].
    lgcn_gather_final<<<batch, EMB, 0, stream>>>(users, uemb, h1, h2, h3, 0,         u_out);
    lgcn_gather_final<<<batch, EMB, 0, stream>>>(pos,   iemb, h1, h2, h3, num_users, p_out);
    lgcn_gather_final<<<batch, EMB, 0, stream>>>(neg,   iemb, h1, h2, h3, num_users, n_out);
}